// GCN_Encoder_68255620268441
// MI455X (gfx1250) — compile-verified
//
#include <hip/hip_runtime.h>
#include <stdint.h>

typedef __attribute__((ext_vector_type(2))) float v2f;
typedef __attribute__((ext_vector_type(8))) float v8f;
typedef __attribute__((ext_vector_type(4))) unsigned int u32x4;
typedef __attribute__((ext_vector_type(4))) int i32x4;
typedef __attribute__((ext_vector_type(8))) int i32x8;

#if defined(__has_builtin)
#if __has_builtin(__builtin_amdgcn_tensor_load_to_lds) && __has_builtin(__builtin_amdgcn_s_wait_tensorcnt)
#define USE_TDM 1
#endif
#endif

// ---------------------------------------------------------------------------
// Small elementwise / setup kernels
// ---------------------------------------------------------------------------
__global__ void fill_f32(float* __restrict__ p, float v, size_t n) {
    size_t i = (size_t)blockIdx.x * blockDim.x + threadIdx.x;
    if (i < n) p[i] = v;
}

// deg initialized to 1.0 (self-loop); add 1 per incoming edge
__global__ void deg_accum(const int* __restrict__ ei, float* __restrict__ deg, int E) {
    int e = blockIdx.x * blockDim.x + threadIdx.x;
    if (e < E) atomicAdd(&deg[ei[E + e]], 1.0f);
}

__global__ void rsqrt_inplace(float* __restrict__ p, int n) {
    int i = blockIdx.x * blockDim.x + threadIdx.x;
    if (i < n) p[i] = rsqrtf(p[i]);   // deg >= 1 always (self-loops)
}

// ---------------------------------------------------------------------------
// BatchNorm (training mode, population variance)
// ---------------------------------------------------------------------------
__global__ void bn_stats(const float* __restrict__ x, float* __restrict__ sums,
                         float* __restrict__ sumsq, int N) {
    const int col = threadIdx.x;            // blockDim.x == 128 == F
    float s = 0.f, s2 = 0.f;
    for (int row = blockIdx.x; row < N; row += gridDim.x) {
        float v = x[(size_t)row * 128 + col];
        s += v; s2 += v * v;
    }
    atomicAdd(&sums[col], s);
    atomicAdd(&sumsq[col], s2);
}

__global__ void bn_finalize(float* __restrict__ stats, const float* __restrict__ gamma,
                            const float* __restrict__ beta, int N) {
    int c = threadIdx.x;                    // 128 threads
    float mu  = stats[c] / (float)N;
    float var = stats[128 + c] / (float)N - mu * mu;
    float rstd = rsqrtf(var + 1e-5f);
    float scale = rstd * gamma[c];
    stats[256 + c] = scale;                 // scale
    stats[384 + c] = beta[c] - mu * scale;  // shift
}

__global__ void bn_apply4(const float4* __restrict__ x, const float* __restrict__ stats,
                          float4* __restrict__ xb, size_t n4) {
    size_t i = (size_t)blockIdx.x * blockDim.x + threadIdx.x;
    if (i >= n4) return;
    int c = (int)((i * 4) & 127);
    float4 v = x[i];
    float4 sc = *(const float4*)(stats + 256 + c);
    float4 sh = *(const float4*)(stats + 384 + c);
    float4 r;
    r.x = v.x * sc.x + sh.x;  r.y = v.y * sc.y + sh.y;
    r.z = v.z * sc.z + sh.z;  r.w = v.w * sc.w + sh.w;
    xb[i] = r;
}

// ---------------------------------------------------------------------------
// WMMA f32 GEMM:  out[N,Nout] (+)= A[N,128] @ W[128,Nout],  Nout = NT*16.
// 256 threads = 8 waves.  W tile (128 x Nout, row-major) is DMA'd into LDS by
// the Tensor Data Mover (TENSOR_LOAD_TO_LDS + s_wait_tensorcnt), issued once
// by wave 0 with a block-uniform D# descriptor (ISA 8.3/8.4).  Each wave
// stages its 16x128 A tile into LDS with coalesced b128 loads (padded stride
// 132 -> conflict-free ds_load_b64 fragments).  Inner loop: LDS reads + NT
// back-to-back V_WMMA_F32_16X16X4_F32.
// ---------------------------------------------------------------------------
#define A_STRIDE 132   // 128 + 4 pad floats

template <int NT>
__global__ void wmma_gemm_f32(const float* __restrict__ A, const float* __restrict__ W,
                              float* __restrict__ out, int nRowTiles, int accumulate) {
    constexpr int K = 128;
    constexpr int Nout = NT * 16;
    __shared__ float lw[K * Nout];              // W, row-major
    __shared__ float la[8 * 16 * A_STRIDE];     // 8 waves x 16x128 A tiles (padded)

    const int tid = threadIdx.x;

    // ---- stage W into LDS ----
#ifdef USE_TDM
    if (tid < 32) {                             // one TDM op per block (wave 0)
        // LDS aperture: low 32 bits of a generic LDS pointer == wave-relative byte offset
        uint32_t ldsW = (uint32_t)(uintptr_t)(&lw[0]);
        uint64_t ga   = (uint64_t)(uintptr_t)W;
        // D# group0: count=1 | lds_addr | global_addr[56:0] | type=2
        u32x4 g0 = { 1u, ldsW, (uint32_t)ga,
                     (uint32_t)((ga >> 32) & 0x01FFFFFFu) | (2u << 30) };
        // D# group1: data_size=4B; tensor_dim0=Nout; tensor_dim1=K;
        //            tile_dim0=Nout; tile_dim1=K; tensor_dim0_stride=Nout
        i32x8 g1 = { (int)(2u << 16),
                     (int)(((uint32_t)Nout & 0xFFFFu) << 16),
                     (int)(((uint32_t)Nout >> 16) | (((uint32_t)K & 0xFFFFu) << 16)),
                     (int)(((uint32_t)K >> 16) | ((uint32_t)Nout << 16)),
                     (int)K,
                     (int)Nout,
                     0, 0 };
        i32x4 gz4 = { 0, 0, 0, 0 };             // 2D tile: groups 2/3 unused
        i32x8 gz8 = { 0, 0, 0, 0, 0, 0, 0, 0 };
        __builtin_amdgcn_tensor_load_to_lds(g0, g1, gz4, gz4, gz8, 0);
        __builtin_amdgcn_s_wait_tensorcnt(0);
    }
#else
    for (int g = tid; g < K * Nout; g += 256) lw[g] = W[g];
#endif
    __syncthreads();

    const int wid = (int)((blockIdx.x * blockDim.x + tid) >> 5);
    if (wid >= nRowTiles) return;               // whole-wave uniform exit
    const int lane = tid & 31;
    const int half = lane >> 4;                 // 0: K=k,k+1   1: K=k+2,k+3
    const int m    = lane & 15;
    const int rowBase = wid * 16;

    // ---- stage this wave's A tile (coalesced b128 loads) ----
    float* myA = la + (tid >> 5) * (16 * A_STRIDE);
    for (int t = lane; t < 16 * 32; t += 32) {  // 512 float4 chunks
        int r  = t >> 5;
        int c4 = (t & 31) * 4;
        float4 v = *(const float4*)(A + (size_t)(rowBase + r) * K + c4);
        *(float4*)(myA + r * A_STRIDE + c4) = v;
    }

    v8f acc[NT];
    if (accumulate) {
        const float* orow = out + (size_t)(rowBase + half * 8) * Nout + m;
#pragma unroll
        for (int ct = 0; ct < NT; ++ct)
#pragma unroll
            for (int r = 0; r < 8; ++r)
                acc[ct][r] = orow[(size_t)r * Nout + ct * 16];
    } else {
#pragma unroll
        for (int ct = 0; ct < NT; ++ct)
#pragma unroll
            for (int r = 0; r < 8; ++r)
                acc[ct][r] = 0.0f;
    }

    const float* afrag = myA + m * A_STRIDE;    // lane's A row
    for (int k = 0; k < K; k += 4) {
        const int kk = k + half * 2;
        v2f a = *(const v2f*)(afrag + kk);
        const float* bbase = lw + (size_t)kk * Nout + m;
        v2f bf[NT];
#pragma unroll
        for (int ct = 0; ct < NT; ++ct) {       // ds_load_2addr_b32 pairs
            const float* bp = bbase + ct * 16;
            v2f b; b.x = bp[0]; b.y = bp[Nout];
            bf[ct] = b;
        }
#pragma unroll
        for (int ct = 0; ct < NT; ++ct)
            acc[ct] = __builtin_amdgcn_wmma_f32_16x16x4_f32(
                false, a, false, bf[ct], (short)0, acc[ct], false, false);
    }

    float* orow = out + (size_t)(rowBase + half * 8) * Nout + m;
#pragma unroll
    for (int ct = 0; ct < NT; ++ct)
#pragma unroll
        for (int r = 0; r < 8; ++r)
            orow[(size_t)r * Nout + ct * 16] = acc[ct][r];
}

// ---------------------------------------------------------------------------
// GCN aggregation:  agg[dst] += h[src] * dinv[src]*dinv[dst]   (atomics)
// One wave per edge; D/32 floats per lane.
// ---------------------------------------------------------------------------
__global__ void init_bias4(float4* __restrict__ agg, const float* __restrict__ bias,
                           size_t n4, int D) {
    size_t i = (size_t)blockIdx.x * blockDim.x + threadIdx.x;
    if (i < n4) agg[i] = *(const float4*)(bias + (int)((i * 4) % D));
}

template <int D>
__global__ void edge_agg(const float* __restrict__ h, const int* __restrict__ ei,
                         const float* __restrict__ dinv, float* __restrict__ agg, int E) {
    constexpr int V = D / 32;
    int warp = (int)((blockIdx.x * blockDim.x + threadIdx.x) >> 5);
    if (warp >= E) return;
    int lane = threadIdx.x & 31;
    int s = ei[warp];
    int d = ei[E + warp];
    float w = dinv[s] * dinv[d];
    const float* hp = h + (size_t)s * D + lane * V;
    float* op = agg + (size_t)d * D + lane * V;
#pragma unroll
    for (int c = 0; c < V; ++c) atomicAdd(op + c, hp[c] * w);
}

// self-loop contribution + optional ReLU (after edge_agg; no atomics needed)
template <int D>
__global__ void node_finish4(const float4* __restrict__ h, const float* __restrict__ dinv,
                             float4* __restrict__ agg, size_t n4, int relu) {
    size_t i = (size_t)blockIdx.x * blockDim.x + threadIdx.x;
    if (i >= n4) return;
    int node = (int)((i * 4) / D);
    float w = dinv[node]; w *= w;
    float4 hv = h[i], av = agg[i], r;
    r.x = av.x + hv.x * w;  r.y = av.y + hv.y * w;
    r.z = av.z + hv.z * w;  r.w = av.w + hv.w * w;
    if (relu) {
        r.x = fmaxf(r.x, 0.f); r.y = fmaxf(r.y, 0.f);
        r.z = fmaxf(r.z, 0.f); r.w = fmaxf(r.w, 0.f);
    }
    agg[i] = r;
}

// ---------------------------------------------------------------------------
extern "C" void kernel_launch(void* const* d_in, const int* in_sizes, int n_in,
                              void* d_out, int out_size, void* d_ws, size_t ws_size,
                              hipStream_t stream) {
    const int F = 128;
    const int N = in_sizes[0] / F;
    const int E = in_sizes[2] / 2;

    const float* x     = (const float*)d_in[0];
    const float* noise = (const float*)d_in[1];
    const int*   ei    = (const int*)  d_in[2];
    const float* gamma = (const float*)d_in[3];
    const float* beta  = (const float*)d_in[4];
    const float* W1    = (const float*)d_in[5];
    const float* b1    = (const float*)d_in[6];
    const float* W2    = (const float*)d_in[7];
    const float* b2    = (const float*)d_in[8];
    const float* Wm    = (const float*)d_in[9];
    const float* bm    = (const float*)d_in[10];
    const float* Ws    = (const float*)d_in[11];
    const float* bs    = (const float*)d_in[12];
    float* out = (float*)d_out;

    // workspace layout (floats)
    float* wsf   = (float*)d_ws;
    float* dinv  = wsf;                          // N
    float* stats = wsf + N;                      // 512: sums|sumsq|scale|shift
    float* xb    = stats + 512;                  // N*F
    float* bufA  = xb   + (size_t)N * F;         // N*128  (GEMM outputs / messages)
    float* bufB  = bufA + (size_t)N * 128;       // N*128  (aggregated h1 then h2)

    const size_t NF  = (size_t)N * F;
    const size_t N64 = (size_t)N * 64;
    const int gN    = (N + 255) / 256;
    const int gE    = (E + 255) / 256;
    const int gNF4  = (int)((NF / 4 + 255) / 256);
    const int gN644 = (int)((N64 / 4 + 255) / 256);
    const int gEw   = (E + 7) / 8;               // one wave per edge
    const int rowTiles = N / 16;
    const int gGemm = (rowTiles * 32 + 255) / 256;

    // 1) degrees -> dinv
    fill_f32<<<gN, 256, 0, stream>>>(dinv, 1.0f, (size_t)N);
    fill_f32<<<2, 256, 0, stream>>>(stats, 0.0f, 512);
    deg_accum<<<gE, 256, 0, stream>>>(ei, dinv, E);
    rsqrt_inplace<<<gN, 256, 0, stream>>>(dinv, N);

    // 2) batch-norm -> xb
    bn_stats<<<1024, 128, 0, stream>>>(x, stats, stats + 128, N);
    bn_finalize<<<1, 128, 0, stream>>>(stats, gamma, beta, N);
    bn_apply4<<<gNF4, 256, 0, stream>>>((const float4*)x, stats, (float4*)xb, NF / 4);

    // 3) layer 1: t = xb@W1 ; h1 = relu(agg(t) + b1)            -> bufB
    wmma_gemm_f32<8><<<gGemm, 256, 0, stream>>>(xb, W1, bufA, rowTiles, 0);
    init_bias4<<<gNF4, 256, 0, stream>>>((float4*)bufB, b1, NF / 4, 128);
    edge_agg<128><<<gEw, 256, 0, stream>>>(bufA, ei, dinv, bufB, E);
    node_finish4<128><<<gNF4, 256, 0, stream>>>((const float4*)bufA, dinv, (float4*)bufB, NF / 4, 1);

    // 4) layer 2: t = [xb|noise|h1]@W2 via 3 accumulating GEMMs; h2 -> bufB
    wmma_gemm_f32<8><<<gGemm, 256, 0, stream>>>(xb,    W2,                 bufA, rowTiles, 0);
    wmma_gemm_f32<8><<<gGemm, 256, 0, stream>>>(noise, W2 + 128 * 128,     bufA, rowTiles, 1);
    wmma_gemm_f32<8><<<gGemm, 256, 0, stream>>>(bufB,  W2 + 2 * 128 * 128, bufA, rowTiles, 1);
    init_bias4<<<gNF4, 256, 0, stream>>>((float4*)bufB, b2, NF / 4, 128);
    edge_agg<128><<<gEw, 256, 0, stream>>>(bufA, ei, dinv, bufB, E);
    node_finish4<128><<<gNF4, 256, 0, stream>>>((const float4*)bufA, dinv, (float4*)bufB, NF / 4, 1);

    // 5) heads: tm/ts = [xb|h2]@{Wm,Ws} via 2 accumulating GEMMs each
    float* tm = bufA;            // N*64
    float* ts = bufA + N64;      // N*64
    wmma_gemm_f32<4><<<gGemm, 256, 0, stream>>>(xb,   Wm,            tm, rowTiles, 0);
    wmma_gemm_f32<4><<<gGemm, 256, 0, stream>>>(bufB, Wm + 128 * 64, tm, rowTiles, 1);
    wmma_gemm_f32<4><<<gGemm, 256, 0, stream>>>(xb,   Ws,            ts, rowTiles, 0);
    wmma_gemm_f32<4><<<gGemm, 256, 0, stream>>>(bufB, Ws + 128 * 64, ts, rowTiles, 1);

    // 6) final aggregations straight into d_out (z_mean, z_logstd)
    float* zm = out;
    float* zs = out + N64;
    init_bias4<<<gN644, 256, 0, stream>>>((float4*)zm, bm, N64 / 4, 64);
    edge_agg<64><<<gEw, 256, 0, stream>>>(tm, ei, dinv, zm, E);
    node_finish4<64><<<gN644, 256, 0, stream>>>((const float4*)tm, dinv, (float4*)zm, N64 / 4, 0);

    init_bias4<<<gN644, 256, 0, stream>>>((float4*)zs, bs, N64 / 4, 64);
    edge_agg<64><<<gEw, 256, 0, stream>>>(ts, ei, dinv, zs, E);
    node_finish4<64><<<gN644, 256, 0, stream>>>((const float4*)ts, dinv, (float4*)zs, N64 / 4, 0);
}